// Shift_gcn_2637109920123
// MI455X (gfx1250) — compile-verified
//
#include <hip/hip_runtime.h>
#include <math.h>

// Shift-GCN fused block for MI455X (gfx1250, wave32).
// Shapes: x0 (64,64,256,25) f32; W (64,64); b (64); mask (25,64);
// gamma/beta (1600); shift_in/out (1600) int32.
//
// ws layout: y (16384x1600 f32) | part_sum (1024x1600) | part_ssq (1024x1600)
//            | mean (1600) | istd (1600)

#define V_DIM 25
#define C_DIM 64
#define NT_ROWS 16384          // n*t = 64*256
#define ROWS_PER_BLK 16
#define NBLK (NT_ROWS / ROWS_PER_BLK)   // 1024 GEMM blocks
#define NCOL 1600              // V*C
#define BN_EPS 1e-5f

// dynamic LDS partition (floats): xs 64*16*25 | msc 1600 | W 4096 | shift_in 1600
#define XS_FLOATS  (C_DIM * ROWS_PER_BLK * V_DIM)   // 25600
#define LDS_BYTES  ((XS_FLOATS + NCOL + C_DIM * C_DIM + NCOL) * 4)   // 131584

typedef __attribute__((ext_vector_type(2))) float v2f;
typedef __attribute__((ext_vector_type(8))) float v8f;

// One workgroup = 8 wave32 = 256 threads, handles 16 rows (one n, 16 t's)
// across all 25 vertices and all 64 output channels.
// Full x0 slab lives in LDS; each vertex is processed start-to-finish with
// only 4 live v8f accumulators (no scratch spills).
__global__ __launch_bounds__(256) void shiftgcn_gemm_kernel(
    const float* __restrict__ x0, const float* __restrict__ W,
    const float* __restrict__ bias, const float* __restrict__ mask,
    const int* __restrict__ shift_in,
    float* __restrict__ yw, float* __restrict__ psum, float* __restrict__ pssq)
{
  extern __shared__ char smem[];
  float* xs    = (float*)smem;          // [c][t][v] = 64*16*25
  float* msc   = xs + XS_FLOATS;        // tanh(mask)+1
  float* wlds  = msc + NCOL;            // 64x64
  int*   sin_l = (int*)(wlds + C_DIM * C_DIM);

  const int tid  = threadIdx.x;
  const int r0   = blockIdx.x * ROWS_PER_BLK;        // global row base
  const int nIdx = r0 >> 8;                          // / 256
  const int t0   = r0 & 255;

  // x slab: 64 chunks of 400 contiguous floats; 16B-aligned (t0*25 % 4 == 0).
  {
    const float4* src = (const float4*)(x0 + (size_t)nIdx * C_DIM * 256 * V_DIM);
    float4* dst = (float4*)xs;
    const int c_stride4 = 256 * V_DIM / 4;           // float4s per channel
    const int t_off4 = t0 * V_DIM / 4;
    for (int i = tid; i < XS_FLOATS / 4; i += 256) {
      int c = i / 100, rem = i % 100;                // 400 floats = 100 float4
      dst[i] = src[c * c_stride4 + t_off4 + rem];
    }
  }
  for (int i = tid; i < NCOL; i += 256) {
    msc[i]   = tanhf(mask[i]) + 1.0f;
    sin_l[i] = shift_in[i];
  }
  for (int i = tid; i < C_DIM * C_DIM; i += 256) wlds[i] = W[i];
  __syncthreads();

  const int wave = tid >> 5;
  const int lane = tid & 31;
  const int ln   = lane & 15;   // N / M position within half
  const int hi   = lane >> 4;   // which half of the wave

  // vertex assignment: wave w -> v = w, w+8, w+16 (+24 for wave 0)
  const int nv = (wave == 0) ? 4 : 3;

  for (int vi = 0; vi < nv; ++vi) {
    const int v = wave + vi * 8;

    v8f acc[4];                        // one vertex: 4 n-tiles, 32 VGPRs
    #pragma unroll
    for (int nt = 0; nt < 4; ++nt) {
      float bv = bias[nt * 16 + ln];   // bias folded into C init
      #pragma unroll
      for (int r = 0; r < 8; ++r) acc[nt][r] = bv;
    }

    for (int k0 = 0; k0 < C_DIM; k0 += 4) {
      // f32 A 16x4 layout: lanes 0-15 hold K=k0,k0+1; lanes 16-31 K=k0+2,k0+3
      const int ka = k0 + 2 * hi;
      const int i0 = sin_l[v * C_DIM + ka];
      const int i1 = sin_l[v * C_DIM + ka + 1];
      v2f a;
      a[0] = xs[(i0 & 63) * (ROWS_PER_BLK * V_DIM) + ln * V_DIM + (i0 >> 6)]
             * msc[v * C_DIM + ka];
      a[1] = xs[(i1 & 63) * (ROWS_PER_BLK * V_DIM) + ln * V_DIM + (i1 >> 6)]
             * msc[v * C_DIM + ka + 1];
      #pragma unroll
      for (int nt = 0; nt < 4; ++nt) {
        // f32 B 4x16 layout (half-split like A/C): v0 = rows k0/k0+2, v1 = k0+1/k0+3
        v2f bf;
        bf[0] = wlds[(k0 + 2 * hi) * C_DIM + nt * 16 + ln];
        bf[1] = wlds[(k0 + 2 * hi + 1) * C_DIM + nt * 16 + ln];
        acc[nt] = __builtin_amdgcn_wmma_f32_16x16x4_f32(
            false, a, false, bf, (short)0, acc[nt], false, false);
      }
    }

    // Store y and emit deterministic per-block BN partial sums.
    // C/D layout: VGPR r, lane: M = r + hi*8, N = ln -> whole lane is one column.
    #pragma unroll
    for (int nt = 0; nt < 4; ++nt) {
      const int q = v * C_DIM + nt * 16 + ln;
      float s = 0.0f, ss = 0.0f;
      #pragma unroll
      for (int r = 0; r < 8; ++r) {
        float val = acc[nt][r];
        yw[(size_t)(r0 + hi * 8 + r) * NCOL + q] = val;
        s += val;
        ss += val * val;
      }
      // combine the two half-wave contributions (fixed order -> deterministic)
      s  += __shfl_xor(s, 16, 32);
      ss += __shfl_xor(ss, 16, 32);
      if (hi == 0) {
        psum[(size_t)blockIdx.x * NCOL + q] = s;
        pssq[(size_t)blockIdx.x * NCOL + q] = ss;
      }
    }
  }
}

// Deterministic reduction of per-block partials -> mean / inv-std per column.
__global__ void bn_stats_kernel(const float* __restrict__ psum,
                                const float* __restrict__ pssq,
                                float* __restrict__ mean,
                                float* __restrict__ istd) {
  int q = blockIdx.x * blockDim.x + threadIdx.x;
  if (q < NCOL) {
    float s = 0.0f, ss = 0.0f;
    for (int b = 0; b < NBLK; ++b) {         // fixed order, coalesced across q
      s  += psum[(size_t)b * NCOL + q];
      ss += pssq[(size_t)b * NCOL + q];
    }
    const float inv_n = 1.0f / (float)NT_ROWS;
    float m = s * inv_n;
    float var = ss * inv_n - m * m;
    mean[q] = m;
    istd[q] = rsqrtf(var + BN_EPS);
  }
}

// One block per row (n,t). shift_out gather stays inside the contiguous
// 6400 B y-row; output written v-fastest so out/x0 accesses are contiguous.
__global__ __launch_bounds__(256) void finalize_kernel(
    const float* __restrict__ yw, const float* __restrict__ x0,
    const int* __restrict__ shift_out,
    const float* __restrict__ gamma, const float* __restrict__ beta,
    const float* __restrict__ mean, const float* __restrict__ istd,
    float* __restrict__ out)
{
  const int row = blockIdx.x;       // 0..16383
  const int n = row >> 8;
  const int t = row & 255;
  const float* yrow = yw + (size_t)row * NCOL;
  for (int jj = threadIdx.x; jj < NCOL; jj += 256) {
    int d = jj / V_DIM;             // v fastest -> coalesced out/x0
    int v = jj % V_DIM;
    int j = v * C_DIM + d;          // column index in z
    int p = shift_out[j];           // column of y feeding z[:, j]
    float val = (yrow[p] - mean[p]) * istd[p] * gamma[j] + beta[j];
    size_t idx = ((size_t)(n * C_DIM + d) * 256 + t) * V_DIM + v;
    float r = val + x0[idx];
    out[idx] = r > 0.0f ? r : 0.0f;
  }
}

extern "C" void kernel_launch(void* const* d_in, const int* in_sizes, int n_in,
                              void* d_out, int out_size, void* d_ws, size_t ws_size,
                              hipStream_t stream) {
  const float* x0    = (const float*)d_in[0];
  const float* W     = (const float*)d_in[1];
  const float* b     = (const float*)d_in[2];
  const float* mask  = (const float*)d_in[3];
  const float* gamma = (const float*)d_in[4];
  const float* beta  = (const float*)d_in[5];
  const int*   sin_p = (const int*)d_in[6];
  const int*   sout  = (const int*)d_in[7];
  float* out = (float*)d_out;

  float* yw   = (float*)d_ws;                       // 16384*1600 f32
  float* psum = yw + (size_t)NT_ROWS * NCOL;        // 1024*1600
  float* pssq = psum + (size_t)NBLK * NCOL;         // 1024*1600
  float* mean = pssq + (size_t)NBLK * NCOL;         // 1600
  float* istd = mean + NCOL;                        // 1600

  // Allow >48KB dynamic LDS (gfx1250: up to 320KB per workgroup).
  (void)hipFuncSetAttribute((const void*)shiftgcn_gemm_kernel,
                            hipFuncAttributeMaxDynamicSharedMemorySize,
                            LDS_BYTES);

  shiftgcn_gemm_kernel<<<NBLK, 256, LDS_BYTES, stream>>>(
      x0, W, b, mask, sin_p, yw, psum, pssq);
  bn_stats_kernel<<<(NCOL + 255) / 256, 256, 0, stream>>>(psum, pssq, mean, istd);
  finalize_kernel<<<NT_ROWS, 256, 0, stream>>>(
      yw, x0, sout, gamma, beta, mean, istd, out);
}